// SecondaryCapsules_1125281432312
// MI455X (gfx1250) — compile-verified
//
#include <hip/hip_runtime.h>

// CapsNet dynamic routing, MI455X / gfx1250.
// K1: u_hat[b,n,o,j] = sum_i x[b,n,i] * W[n,o,i,j]   via V_WMMA_F32_16X16X4_F32
// K2: 3 rounds of routing (softmax -> s -> squash -> agreement), u_hat streamed
//     from workspace (189 MB, ~L2-resident), b_ij/c_ij kept in LDS per block.

typedef float v2f __attribute__((ext_vector_type(2)));
typedef float v8f __attribute__((ext_vector_type(8)));

#define B_SZ   256
#define NIN    1152
#define NOUT   10
#define DIN    8
#define DOUT   16
#define OJ     (NOUT * DOUT)      // 160
#define UHAT_PER_B ((size_t)NIN * OJ)  // 184320 floats per batch element

// ---------------------------------------------------------------------------
// Kernel 1: u_hat GEMM with fp32 WMMA (16x16x4, two K-steps for K=8).
// One wave handles one (n, 16-batch-row tile) unit; 4 waves per block.
// Grid: 1152 * 16 / 4 = 4608 blocks of 128 threads.
// ---------------------------------------------------------------------------
__global__ __launch_bounds__(128) void uhat_wmma_kernel(
    const float* __restrict__ x,     // [256,1152,8]
    const float* __restrict__ W,     // [1152,10,8,16]
    float* __restrict__ uhat)        // [256,1152,10,16]
{
  const int wave = (blockIdx.x << 2) + (threadIdx.x >> 5);
  const int lane = threadIdx.x & 31;
  const int n    = wave >> 4;        // 0..1151
  const int bt   = wave & 15;        // 0..15
  const int half = lane >> 4;        // lane-half (0: lanes 0-15, 1: lanes 16-31)
  const int l    = lane & 15;
  const int b0   = bt << 4;

  // A-matrix 16x4 fp32 layout (ISA 7.12.2): lane l row M=l; lanes 0-15 carry
  // K = {k0, k0+1} in v[0..1], lanes 16-31 carry K = {k0+2, k0+3}.
  const float* xrow = x + (size_t)(b0 + l) * (NIN * DIN) + n * DIN + 2 * half;
  const v2f A0 = *(const v2f*)(xrow);       // K-step k0 = 0   (8B aligned)
  const v2f A1 = *(const v2f*)(xrow + 4);   // K-step k0 = 4

  // B-matrix 4x16 fp32 layout: VGPR v, lane-half h -> row K = k0 + v + 2h,
  // column N = l. With d_out = 16, each 16-wide (o,j) tile is exactly one o.
  const float* wbase = W + (size_t)n * (NOUT * DIN * DOUT) + l;

  #pragma unroll
  for (int o = 0; o < NOUT; ++o) {
    const float* wp = wbase + o * (DIN * DOUT);
    v2f B0, B1;
    B0.x = wp[(0 + 2 * half) * DOUT];
    B0.y = wp[(1 + 2 * half) * DOUT];
    B1.x = wp[(4 + 2 * half) * DOUT];
    B1.y = wp[(5 + 2 * half) * DOUT];

    v8f c = {};
    // 8 args: (neg_a, A, neg_b, B, c_mod, C, reuse_a, reuse_b)
    c = __builtin_amdgcn_wmma_f32_16x16x4_f32(false, A0, false, B0,
                                              (short)0, c, false, false);
    c = __builtin_amdgcn_wmma_f32_16x16x4_f32(false, A1, false, B1,
                                              (short)0, c, false, false);

    // C/D layout: VGPR v, lane-half h -> M = v + 8h (batch row), N = l (j).
    size_t outBase =
        (((size_t)(b0 + 8 * half) * NIN + n) * NOUT + o) * DOUT + l;
    #pragma unroll
    for (int v = 0; v < 8; ++v)
      uhat[outBase + (size_t)v * UHAT_PER_B] = c[v];
  }
}

// ---------------------------------------------------------------------------
// Kernel 2: routing. One block per batch element. b_ij / c_ij live in LDS.
// ---------------------------------------------------------------------------
__global__ __launch_bounds__(256) void routing_kernel(
    const float* __restrict__ uhat,  // [256,1152,10,16]
    float* __restrict__ out)         // [256,10,16]
{
  __shared__ float b_lds[NIN * NOUT];     // 46080 B
  __shared__ float c_lds[NIN * NOUT];     // 46080 B
  __shared__ float sv_lds[OJ];            // s_j, then v_j in place
  __shared__ float scale_lds[NOUT];

  const int b = blockIdx.x;
  const int t = threadIdx.x;
  const float* ub = uhat + (size_t)b * UHAT_PER_B;

  for (int i = t; i < NIN * NOUT; i += 256) b_lds[i] = 0.0f;
  __syncthreads();

  for (int r = 0; r < 3; ++r) {
    // Phase A: c[n][:] = softmax(b[n][:]) over o   (n-parallel)
    for (int n = t; n < NIN; n += 256) {
      float bv[NOUT];
      float m = -1e30f;
      #pragma unroll
      for (int o = 0; o < NOUT; ++o) {
        bv[o] = b_lds[n * NOUT + o];
        m = fmaxf(m, bv[o]);
      }
      float den = 0.0f;
      #pragma unroll
      for (int o = 0; o < NOUT; ++o) {
        bv[o] = __expf(bv[o] - m);
        den += bv[o];
      }
      const float inv = 1.0f / den;
      #pragma unroll
      for (int o = 0; o < NOUT; ++o) c_lds[n * NOUT + o] = bv[o] * inv;
    }
    __syncthreads();

    // Phase B: s[o][j] = sum_n c[n][o] * u_hat[n][o][j]   ((o,j)-parallel)
    if (t < OJ) {
      const int o = t >> 4;
      const float* up = ub + t;            // element (o, j) of row n=0
      float a0 = 0.f, a1 = 0.f, a2 = 0.f, a3 = 0.f;
      for (int n = 0; n < NIN; n += 4) {
        a0 += c_lds[(n + 0) * NOUT + o] * up[(size_t)(n + 0) * OJ];
        a1 += c_lds[(n + 1) * NOUT + o] * up[(size_t)(n + 1) * OJ];
        a2 += c_lds[(n + 2) * NOUT + o] * up[(size_t)(n + 2) * OJ];
        a3 += c_lds[(n + 3) * NOUT + o] * up[(size_t)(n + 3) * OJ];
      }
      sv_lds[t] = (a0 + a1) + (a2 + a3);
    }
    __syncthreads();

    // Phase C: squash scale per o, then v = scale * s
    if (t < NOUT) {
      float ns = 0.0f;
      #pragma unroll
      for (int j = 0; j < DOUT; ++j) {
        const float s = sv_lds[t * DOUT + j];
        ns += s * s;
      }
      const float norm = sqrtf(ns + 1e-9f);
      scale_lds[t] = ns / (1.0f + ns) / norm;
    }
    __syncthreads();
    if (t < OJ) sv_lds[t] *= scale_lds[t >> 4];
    __syncthreads();

    // Phase D: b[n][o] += sum_j u_hat[n][o][j] * v[o][j]   (n-parallel)
    if (r < 2) {
      for (int n = t; n < NIN; n += 256) {
        const float* up = ub + (size_t)n * OJ;
        #pragma unroll
        for (int o = 0; o < NOUT; ++o) {
          float acc = 0.0f;
          #pragma unroll
          for (int j = 0; j < DOUT; j += 4) {
            const float4 u4 = *(const float4*)(up + o * DOUT + j);
            acc += u4.x * sv_lds[o * DOUT + j + 0]
                 + u4.y * sv_lds[o * DOUT + j + 1]
                 + u4.z * sv_lds[o * DOUT + j + 2]
                 + u4.w * sv_lds[o * DOUT + j + 3];
          }
          b_lds[n * NOUT + o] += acc;
        }
      }
      __syncthreads();
    }
  }

  if (t < OJ) out[b * OJ + t] = sv_lds[t];
}

// ---------------------------------------------------------------------------
extern "C" void kernel_launch(void* const* d_in, const int* in_sizes, int n_in,
                              void* d_out, int out_size, void* d_ws,
                              size_t ws_size, hipStream_t stream) {
  const float* x = (const float*)d_in[0];   // [256,1152,8]
  const float* W = (const float*)d_in[1];   // [1152,10,8,16]
  float* uhat = (float*)d_ws;               // 47,185,920 floats (189 MB)
  float* out  = (float*)d_out;              // [256,10,16]

  (void)in_sizes; (void)n_in; (void)out_size; (void)ws_size;

  // 1152 n-values * 16 batch tiles, 4 wave-units per 128-thread block.
  uhat_wmma_kernel<<<dim3((NIN * 16) / 4), dim3(128), 0, stream>>>(x, W, uhat);
  routing_kernel<<<dim3(B_SZ), dim3(256), 0, stream>>>(uhat, out);
}